// NACSysGrid_18365280157979
// MI455X (gfx1250) — compile-verified
//
#include <hip/hip_runtime.h>
#include <stdint.h>

// ---------------------------------------------------------------------------
// NACSysGrid: 1024x1024 CNN (locally-connected 3x3 stencil) integrated with
// fixed-step Tsit5 for 32 steps. Bandwidth-bound; per-edge weights indexed in
// closed form (the reference builds edges in 8 dense (dr,dc) blocks).
// CDNA5 path: LDS tiling with GLOBAL_LOAD_ASYNC_TO_LDS (ASYNCcnt) staging.
// ---------------------------------------------------------------------------

#define NR 1024
#define NC 1024
#define NN (NR * NC)
#define TS 32          // output tile = 32x32 per block
#define LDSW 35        // padded LDS row stride for 34 columns

// Edge-block layout: blocks in loop order dr=-1..1, dc=-1..1 (skip 0,0).
// Block d covers sources (jr,jc) with jr in [r0, r0+H), jc in [c0, c0+W),
// within-block index = (jr-r0)*W + (jc-c0).
__device__ constexpr int DIRS[8][2] = {
    {-1,-1},{-1,0},{-1,1},{0,-1},{0,1},{1,-1},{1,0},{1,1}};
__device__ constexpr int KBASE[8] = {
    0, 1046529, 2094081, 3140610, 4188162, 5235714, 6282243, 7329795};

// Tsit5 tableau (float)
constexpr float fDT  = 0.03125f;
constexpr float fA21 = (float)0.161;
constexpr float fA31 = (float)-0.008480655492356989, fA32 = (float)0.335480655492357;
constexpr float fA41 = (float)2.8971530571054935,  fA42 = (float)-6.359448489975075,  fA43 = (float)4.3622954328695815;
constexpr float fA51 = (float)5.325864828439257,   fA52 = (float)-11.748883564062828, fA53 = (float)7.4955393428898365, fA54 = (float)-0.09249506636175525;
constexpr float fA61 = (float)5.86145544294642,    fA62 = (float)-12.92096931784711,  fA63 = (float)8.159367898576159,  fA64 = (float)-0.071584973281401, fA65 = (float)-0.028269050394068383;
constexpr float fB1  = (float)0.09646076681806523, fB2  = (float)0.01,                fB3  = (float)0.4798896504144996;
constexpr float fB4  = (float)1.379008574103742,   fB5  = (float)-3.290069515436081,  fB6  = (float)2.324710524099774;

// ---- async global -> LDS staging (gfx1250), guarded with fallback ---------
#if __has_builtin(__builtin_amdgcn_global_load_async_to_lds_b32) && \
    __has_builtin(__builtin_amdgcn_s_wait_asynccnt)
#define USE_ASYNC_LDS 1
#endif

typedef __attribute__((address_space(1))) int g_int;
typedef __attribute__((address_space(3))) int l_int;

__device__ __forceinline__ void g2l_b32(const float* g, float* l) {
#ifdef USE_ASYNC_LDS
    __builtin_amdgcn_global_load_async_to_lds_b32(
        (g_int*)(const_cast<float*>(g)),
        (l_int*)(l),
        0, 0);
#else
    *l = *g;
#endif
}

__device__ __forceinline__ void g2l_wait() {
#ifdef USE_ASYNC_LDS
    __builtin_amdgcn_s_wait_asynccnt(0);
#endif
}

__device__ __forceinline__ float clip1(float v) {
    return fminf(fmaxf(v, -1.0f), 1.0f);
}

// ---------------------------------------------------------------------------
// One RK stage: reads stage input u (tile+halo via LDS), computes
// f = -u + z + sum_d k_d * clip(u_neighbor), then per-node combo:
//  S<6 : kout = f ; unext = y + DT * (a_s1 k1 + ... + a_ss f)
//  S==6: unext(=y) = y + DT*(B1 k1+..+B5 k5+B6 f); optional clipped d_out
// ---------------------------------------------------------------------------
template <int S>
__global__ __launch_bounds__(256) void stage_kernel(
    const float* __restrict__ u,
    const float* ybase,
    const float* __restrict__ z,
    const float* __restrict__ kk,
    const float* __restrict__ k1, const float* __restrict__ k2,
    const float* __restrict__ k3, const float* __restrict__ k4,
    const float* __restrict__ k5,
    float* __restrict__ kout,
    float* unext,
    float* out, int write_out)
{
    __shared__ float tile[34 * LDSW];

    const int bc = blockIdx.x * TS;
    const int br = blockIdx.y * TS;
    const int t  = threadIdx.y * 32 + threadIdx.x;   // 0..255

    // Stage-input tile + 1 halo into LDS (clamped at grid borders; border
    // contributions are masked out later, so clamped values never matter).
    for (int e = t; e < 34 * 34; e += 256) {
        const int lr = e / 34;
        const int lc = e - lr * 34;
        int gr = br - 1 + lr; gr = gr < 0 ? 0 : (gr > NR - 1 ? NR - 1 : gr);
        int gc = bc - 1 + lc; gc = gc < 0 ? 0 : (gc > NC - 1 ? NC - 1 : gc);
        g2l_b32(u + gr * NC + gc, &tile[lr * LDSW + lc]);
    }
    g2l_wait();
    __syncthreads();

    #pragma unroll
    for (int q = 0; q < 4; ++q) {
        const int lr  = threadIdx.y + 8 * q;
        const int lc  = threadIdx.x;
        const int ri  = br + lr;
        const int ci  = bc + lc;
        const int idx = ri * NC + ci;

        const float ui = tile[(lr + 1) * LDSW + (lc + 1)];
        float acc = z[idx] - ui;

        #pragma unroll
        for (int d = 0; d < 8; ++d) {
            const int dr = DIRS[d][0], dc = DIRS[d][1];
            const int jr = ri - dr, jc = ci - dc;          // source node
            const bool ok = ((unsigned)jr < (unsigned)NR) &&
                            ((unsigned)jc < (unsigned)NC);
            const int W  = NC - (dc < 0 ? -dc : dc);
            const int r0 = dr < 0 ? -dr : 0;
            const int c0 = dc < 0 ? -dc : 0;
            int kidx = KBASE[d] + (jr - r0) * W + (jc - c0);
            kidx = ok ? kidx : 0;                           // safe dummy load
            const float kv = ok ? kk[kidx] : 0.0f;
            const float nv = clip1(tile[(lr + 1 - dr) * LDSW + (lc + 1 - dc)]);
            acc = fmaf(kv, nv, acc);
        }
        const float f = acc;

        if constexpr (S == 1) {
            kout[idx]  = f;                 // u == y at stage 1
            unext[idx] = ui + fDT * (fA21 * f);
        } else if constexpr (S == 2) {
            const float yb = ybase[idx];
            kout[idx]  = f;
            unext[idx] = yb + fDT * (fA31 * k1[idx] + fA32 * f);
        } else if constexpr (S == 3) {
            const float yb = ybase[idx];
            kout[idx]  = f;
            unext[idx] = yb + fDT * (fA41 * k1[idx] + fA42 * k2[idx] + fA43 * f);
        } else if constexpr (S == 4) {
            const float yb = ybase[idx];
            kout[idx]  = f;
            unext[idx] = yb + fDT * (fA51 * k1[idx] + fA52 * k2[idx] +
                                     fA53 * k3[idx] + fA54 * f);
        } else if constexpr (S == 5) {
            const float yb = ybase[idx];
            kout[idx]  = f;
            unext[idx] = yb + fDT * (fA61 * k1[idx] + fA62 * k2[idx] +
                                     fA63 * k3[idx] + fA64 * k4[idx] + fA65 * f);
        } else {                             // S == 6: advance the state
            const float yb = ybase[idx];
            const float yn = yb + fDT * (fB1 * k1[idx] + fB2 * k2[idx] +
                                         fB3 * k3[idx] + fB4 * k4[idx] +
                                         fB5 * k5[idx] + fB6 * f);
            unext[idx] = yn;                 // unext == y buffer (in place)
            if (write_out) out[idx] = clip1(yn);
        }
    }
}

// ---------------------------------------------------------------------------
// Workspace layout (floats): y, uA, uB, k1..k5  -> 8 * NN * 4 B = 32 MB.
// ---------------------------------------------------------------------------
extern "C" void kernel_launch(void* const* d_in, const int* in_sizes, int n_in,
                              void* d_out, int out_size, void* d_ws, size_t ws_size,
                              hipStream_t stream) {
    (void)in_sizes; (void)n_in; (void)out_size; (void)ws_size;

    const float* x  = (const float*)d_in[0];   // initial state, 1024x1024
    const float* z  = (const float*)d_in[1];   // per-node bias, N
    const float* kk = (const float*)d_in[2];   // per-edge weights, 8376324
    // d_in[3]=src, d_in[4]=dst are unused: edge indexing is closed-form.

    float* ws = (float*)d_ws;
    float* y  = ws + (size_t)0 * NN;
    float* uA = ws + (size_t)1 * NN;
    float* uB = ws + (size_t)2 * NN;
    float* k1 = ws + (size_t)3 * NN;
    float* k2 = ws + (size_t)4 * NN;
    float* k3 = ws + (size_t)5 * NN;
    float* k4 = ws + (size_t)6 * NN;
    float* k5 = ws + (size_t)7 * NN;
    float* out = (float*)d_out;

    (void)hipMemcpyAsync(y, x, (size_t)NN * sizeof(float),
                         hipMemcpyDeviceToDevice, stream);

    dim3 grid(NC / TS, NR / TS);   // 32 x 32 blocks
    dim3 block(32, 8);             // 256 threads = 8 wave32

    for (int step = 0; step < 32; ++step) {
        const int last = (step == 31) ? 1 : 0;
        stage_kernel<1><<<grid, block, 0, stream>>>(y,  y, z, kk, k1, k2, k3, k4, k5, k1, uA, out, 0);
        stage_kernel<2><<<grid, block, 0, stream>>>(uA, y, z, kk, k1, k2, k3, k4, k5, k2, uB, out, 0);
        stage_kernel<3><<<grid, block, 0, stream>>>(uB, y, z, kk, k1, k2, k3, k4, k5, k3, uA, out, 0);
        stage_kernel<4><<<grid, block, 0, stream>>>(uA, y, z, kk, k1, k2, k3, k4, k5, k4, uB, out, 0);
        stage_kernel<5><<<grid, block, 0, stream>>>(uB, y, z, kk, k1, k2, k3, k4, k5, k5, uA, out, 0);
        stage_kernel<6><<<grid, block, 0, stream>>>(uA, y, z, kk, k1, k2, k3, k4, k5, nullptr, y, out, last);
    }
}